// RGCNLayer_32736240730905
// MI455X (gfx1250) — compile-verified
//
#include <hip/hip_runtime.h>

typedef __attribute__((ext_vector_type(2))) float v2f;
typedef __attribute__((ext_vector_type(8))) float v8f;

// Problem dims from the reference setup_inputs()
constexpr int B_ = 4, N_ = 10000, F_ = 128, U_ = 128, R_ = 4, E_ = 160000;
constexpr int M_ = B_ * N_;          // 40000 rows in the flattened GEMM
constexpr int TOTAL_OUT = B_ * N_ * U_;  // 5.12M floats

// out[b,n,u] = bias[u]  (accumulation target for the atomic scatter)
__global__ void init_bias_kernel(float* __restrict__ out,
                                 const float* __restrict__ bias, int total) {
    int i = blockIdx.x * blockDim.x + threadIdx.x;
    if (i < total) out[i] = bias[i & (U_ - 1)];
}

__global__ void relu_kernel(float* __restrict__ out, int total) {
    int i = blockIdx.x * blockDim.x + threadIdx.x;
    if (i < total) out[i] = fmaxf(out[i], 0.0f);
}

// Y[M,U] = X[M,F] @ Wr[F,U] using V_WMMA_F32_16X16X4_F32.
// One wave owns one 16x16 tile of Y; K=128 consumed 4 at a time.
// grid = (M/16, U/64), block = 128 threads (4 waves -> 4 column tiles).
__global__ void gemm_xw_kernel(const float* __restrict__ X,
                               const float* __restrict__ Wr,
                               float* __restrict__ Y) {
    const int lane  = threadIdx.x & 31;
    const int wave  = threadIdx.x >> 5;
    const int mtile = blockIdx.x;
    const int ctile = blockIdx.y * 4 + wave;
    const int l16   = lane & 15;
    const int hi    = lane >> 4;          // 0: lanes 0-15, 1: lanes 16-31
    const int koff  = hi << 1;            // A/B VGPR0 holds K=koff, VGPR1 K=koff+1

    const float* xrow = X  + (size_t)(mtile * 16 + l16) * F_;   // A: lane = row M
    const float* wcol = Wr + ctile * 16 + l16;                  // B: lane = col N

    v8f acc = {};
#pragma unroll 8
    for (int k = 0; k < F_; k += 4) {
        v2f a, bm;
        a.x  = xrow[k + koff];
        a.y  = xrow[k + koff + 1];
        bm.x = wcol[(size_t)(k + koff)     * U_];
        bm.y = wcol[(size_t)(k + koff + 1) * U_];
        acc = __builtin_amdgcn_wmma_f32_16x16x4_f32(
            /*neg_a=*/false, a, /*neg_b=*/false, bm,
            /*c_mod=*/(short)0, acc, /*reuse_a=*/false, /*reuse_b=*/false);
    }

    // C/D layout: VGPR i, lanes 0-15 -> M=i, lanes 16-31 -> M=i+8; N = lane%16
    float* yt = Y + (size_t)(mtile * 16 + (hi << 3)) * U_ + ctile * 16 + l16;
#pragma unroll
    for (int i = 0; i < 8; ++i) yt[(size_t)i * U_] = acc[i];
}

// One wave per (b, edge): out[b,dst,:] += v * Y[b,src,:]
// Each lane covers u = lane + {0,32,64,96}: 4 hardware f32 atomics per lane.
// grid = (E/8, B), block = 256 (8 waves).
__global__ void scatter_kernel(const float* __restrict__ Y,
                               float* __restrict__ out,
                               const float* __restrict__ ev,
                               const int* __restrict__ src,
                               const int* __restrict__ dst) {
    const int lane = threadIdx.x & 31;
    const int wave = threadIdx.x >> 5;
    const int e    = blockIdx.x * 8 + wave;
    const int b    = blockIdx.y;

    const int   s = src[e];
    const int   d = dst[e];
    const float v = ev[e];

    const float* yrow = Y   + (size_t)(b * N_ + s) * U_;
    float*       orow = out + (size_t)(b * N_ + d) * U_;
#pragma unroll
    for (int j = 0; j < 4; ++j) {
        const int u = lane + 32 * j;
        unsafeAtomicAdd(&orow[u], v * yrow[u]);   // global_atomic_add_f32, no return
    }
}

extern "C" void kernel_launch(void* const* d_in, const int* in_sizes, int n_in,
                              void* d_out, int out_size, void* d_ws, size_t ws_size,
                              hipStream_t stream) {
    const float* x    = (const float*)d_in[0];   // [B,N,F]
    const float* ev   = (const float*)d_in[1];   // [R,E]
    const float* W    = (const float*)d_in[2];   // [R,F,U]
    const float* bias = (const float*)d_in[3];   // [U]
    const int*   src  = (const int*)d_in[4];     // [R,E]
    const int*   dst  = (const int*)d_in[5];     // [R,E]
    float*       out  = (float*)d_out;           // [B,N,U]
    float*       y    = (float*)d_ws;            // scratch [B*N, U] = 20.48 MB

    init_bias_kernel<<<dim3((TOTAL_OUT + 255) / 256), dim3(256), 0, stream>>>(out, bias, TOTAL_OUT);

    for (int r = 0; r < R_; ++r) {
        gemm_xw_kernel<<<dim3(M_ / 16, U_ / 64), dim3(128), 0, stream>>>(
            x, W + (size_t)r * F_ * U_, y);
        scatter_kernel<<<dim3(E_ / 8, B_), dim3(256), 0, stream>>>(
            y, out, ev + (size_t)r * E_, src + (size_t)r * E_, dst + (size_t)r * E_);
    }

    relu_kernel<<<dim3((TOTAL_OUT + 255) / 256), dim3(256), 0, stream>>>(out, TOTAL_OUT);
}